// VectorQuantizerEMA_41489384079392
// MI455X (gfx1250) — compile-verified
//
#include <hip/hip_runtime.h>
#include <math.h>

typedef __attribute__((ext_vector_type(2))) float v2f;
typedef __attribute__((ext_vector_type(8))) float v8f;

#define NROWS 32768          // B*H*W = 32*32*32
#define DDIM  256            // embedding dim (== C)
#define KCODES 1024          // num embeddings
#define NELEM (NROWS * DDIM) // 8388608
#define LOSS_BLOCKS 4096
#define EPS 1e-5f

// Branch-free argmin update: strict-less wins; equal value keeps smaller index
// (matches jnp.argmin first-occurrence semantics). Non-short-circuit ops so
// the compiler emits v_cmp + v_cndmask instead of exec-mask control flow.
__device__ __forceinline__ void amin_update(float& bv, int& bi, float dst, int col) {
    const bool take = (dst < bv) | ((dst == bv) & (col < bi));
    bv = take ? dst : bv;
    bi = take ? col : bi;
}

// ---------------------------------------------------------------------------
// Kernel 0: enorm[k] = sum_d E[d][k]^2   (E is row-major (D,K))
// ---------------------------------------------------------------------------
__global__ __launch_bounds__(256) void vq_enorm_kernel(const float* __restrict__ E,
                                                       float* __restrict__ enorm) {
    int k = blockIdx.x * 256 + threadIdx.x;
    if (k < KCODES) {
        float s = 0.f;
#pragma unroll 8
        for (int d = 0; d < DDIM; ++d) {
            float v = E[d * KCODES + k];
            s += v * v;
        }
        enorm[k] = s;
    }
}

// ---------------------------------------------------------------------------
// Kernel 1: fused  P = X @ E  (fp32 WMMA 16x16x4)  +  argmin_k(enorm[k]-2P)
// Block = 256 threads (8 waves). Block owns 32 rows of X (two 16-row strips
// staged in LDS). Each wave holds acc[2 M-strips][4 N-tiles] so each B
// fragment (L2 load) feeds 2 WMMAs and each A fragment (ds load) feeds 4.
// Outer loop = 2 iters covers K=1024 codes.
// ---------------------------------------------------------------------------
__global__ __launch_bounds__(256) void vq_argmin_gemm_kernel(
    const float* __restrict__ X, const float* __restrict__ E,
    const float* __restrict__ enorm, int* __restrict__ idx_out,
    float* __restrict__ idx_f_out) {
    __shared__ float xs[32][260];   // 32 rows x 256 cols, pad -> conflict-free
    __shared__ float bv_s[8][32];
    __shared__ int   bi_s[8][32];

    const int tid  = threadIdx.x;
    const int wave = tid >> 5;
    const int lane = tid & 31;
    const int m0   = blockIdx.x * 32;

    // Stage 32x256 strip of X into LDS (coalesced)
    for (int t = tid; t < 32 * 256; t += 256) {
        int r = t >> 8, c = t & 255;
        xs[r][c] = X[(m0 + r) * DDIM + c];
    }
    __syncthreads();

    const int half = lane >> 4;     // 0: K=0,1 ; 1: K=2,3 (A/B ISA layout)
    const int l16  = lane & 15;
    const int koff = half * 2;

    float bestv[2][8];
    int   besti[2][8];
#pragma unroll
    for (int s = 0; s < 2; ++s)
#pragma unroll
        for (int j = 0; j < 8; ++j) { bestv[s][j] = __builtin_inff(); besti[s][j] = 0x7fffffff; }

#pragma unroll 1
    for (int outer = 0; outer < 2; ++outer) {
        const int nbase = outer * 512 + wave * 16;  // tiles at nbase + t*128
        v8f acc[2][4];
#pragma unroll
        for (int s = 0; s < 2; ++s)
#pragma unroll
            for (int t = 0; t < 4; ++t)
                acc[s][t] = (v8f){0.f, 0.f, 0.f, 0.f, 0.f, 0.f, 0.f, 0.f};

        const float* ecol = E + nbase + l16;        // coalesced over 16 lanes
#pragma unroll 2
        for (int k0 = 0; k0 < DDIM; k0 += 4) {
            // A 16x4 per strip: lane(l16,half) -> X[m0+s*16+l16][k0+koff +0/1]
            const float2 ax0 = *(const float2*)&xs[l16][k0 + koff];
            const float2 ax1 = *(const float2*)&xs[16 + l16][k0 + koff];
            v2f a0; a0.x = ax0.x; a0.y = ax0.y;
            v2f a1; a1.x = ax1.x; a1.y = ax1.y;
            // B 4x16 per N-tile: lane(l16,half) -> E[k0+koff +0/1][n0+l16]
            v2f b[4];
#pragma unroll
            for (int t = 0; t < 4; ++t) {
                b[t].x = ecol[(k0 + koff) * KCODES + t * 128];
                b[t].y = ecol[(k0 + koff + 1) * KCODES + t * 128];
            }
#pragma unroll
            for (int t = 0; t < 4; ++t) {
                acc[0][t] = __builtin_amdgcn_wmma_f32_16x16x4_f32(
                    false, a0, false, b[t], (short)0, acc[0][t], false, false);
                acc[1][t] = __builtin_amdgcn_wmma_f32_16x16x4_f32(
                    false, a1, false, b[t], (short)0, acc[1][t], false, false);
            }
        }

        // C layout: lane l16 = column n0+l16 ; VGPR j = row (half*8 + j)
#pragma unroll
        for (int t = 0; t < 4; ++t) {
            const int   col = nbase + t * 128 + l16;
            const float en  = enorm[col];
#pragma unroll
            for (int s = 0; s < 2; ++s)
#pragma unroll
            for (int j = 0; j < 8; ++j) {
                amin_update(bestv[s][j], besti[s][j],
                            __builtin_fmaf(-2.0f, acc[s][t][j], en), col);
            }
        }
    }

    // Reduce over the 16 lanes that share a row-set (first-index tie-break)
#pragma unroll
    for (int s = 0; s < 2; ++s)
#pragma unroll
    for (int j = 0; j < 8; ++j) {
        float v = bestv[s][j]; int i = besti[s][j];
#pragma unroll
        for (int off = 8; off >= 1; off >>= 1) {
            float ov = __shfl_xor(v, off, 16);
            int   oi = __shfl_xor(i, off, 16);
            amin_update(v, i, ov, oi);
        }
        bestv[s][j] = v; besti[s][j] = i;
    }
    if (l16 == 0) {
#pragma unroll
        for (int s = 0; s < 2; ++s)
#pragma unroll
        for (int j = 0; j < 8; ++j) {
            int row = s * 16 + half * 8 + j;
            bv_s[wave][row] = bestv[s][j];
            bi_s[wave][row] = besti[s][j];
        }
    }
    __syncthreads();

    // Fixed-order cross-wave reduction (deterministic)
    if (tid < 32) {
        float v = bv_s[0][tid]; int i = bi_s[0][tid];
#pragma unroll
        for (int w = 1; w < 8; ++w) {
            amin_update(v, i, bv_s[w][tid], bi_s[w][tid]);
        }
        idx_out[m0 + tid]   = i;
        idx_f_out[m0 + tid] = (float)i;
    }
}

// ---------------------------------------------------------------------------
// Kernel 2: quantized_ste = x + (E[d][idx[r]] - x); block-partial MSE sums
// ---------------------------------------------------------------------------
__global__ __launch_bounds__(256) void vq_gather_kernel(
    const float* __restrict__ X, const float* __restrict__ E,
    const int* __restrict__ idx, float* __restrict__ out,
    float* __restrict__ part) {
    __shared__ float red[256];
    const int tid    = threadIdx.x;
    const int gid    = blockIdx.x * 256 + tid;
    const int stride = LOSS_BLOCKS * 256;
    float acc = 0.f;
#pragma unroll
    for (int it = 0; it < NELEM / (LOSS_BLOCKS * 256); ++it) {  // 8 iters
        int e = gid + it * stride;
        int r = e >> 8;        // row  (D == 256)
        int d = e & 255;       // dim
        float q = E[d * KCODES + idx[r]];
        float x = X[e];
        float diff = q - x;
        out[e] = x + diff;     // straight-through value
        acc += diff * diff;
    }
    red[tid] = acc;
    __syncthreads();
#pragma unroll
    for (int s = 128; s >= 1; s >>= 1) {
        if (tid < s) red[tid] += red[tid + s];
        __syncthreads();
    }
    if (tid == 0) part[blockIdx.x] = red[0];
}

// ---------------------------------------------------------------------------
// Kernel 3: finalize vq_loss, perplexity, code_usage (single block, fixed order)
// ---------------------------------------------------------------------------
__global__ __launch_bounds__(256) void vq_finalize_kernel(
    const float* __restrict__ part, const float* __restrict__ cs,
    float* __restrict__ out_scalars) {
    __shared__ float red[256];
    const int tid = threadIdx.x;

    // vq_loss = (1 + 0.25) * mean((q-x)^2)
    float a = 0.f;
    for (int i = tid; i < LOSS_BLOCKS; i += 256) a += part[i];
    red[tid] = a; __syncthreads();
    for (int s = 128; s >= 1; s >>= 1) { if (tid < s) red[tid] += red[tid + s]; __syncthreads(); }
    float vq_loss = 1.25f * (red[0] / (float)NELEM);
    __syncthreads();

    // sum(cluster_size)
    float s0 = 0.f;
    for (int i = tid; i < KCODES; i += 256) s0 += cs[i];
    red[tid] = s0; __syncthreads();
    for (int s = 128; s >= 1; s >>= 1) { if (tid < s) red[tid] += red[tid + s]; __syncthreads(); }
    float total = red[0] + EPS;
    __syncthreads();

    // entropy and used-code count
    float h = 0.f, used = 0.f;
    for (int i = tid; i < KCODES; i += 256) {
        float p = cs[i] / total;
        h += p * logf(p + EPS);
        used += (cs[i] > EPS) ? 1.f : 0.f;
    }
    red[tid] = h; __syncthreads();
    for (int s = 128; s >= 1; s >>= 1) { if (tid < s) red[tid] += red[tid + s]; __syncthreads(); }
    float perp = expf(-red[0]);
    __syncthreads();
    red[tid] = used; __syncthreads();
    for (int s = 128; s >= 1; s >>= 1) { if (tid < s) red[tid] += red[tid + s]; __syncthreads(); }
    float rate = red[0] / (float)KCODES;

    if (tid == 0) {
        out_scalars[0] = vq_loss;
        out_scalars[1] = perp;
        out_scalars[2] = rate;
    }
}

// ---------------------------------------------------------------------------
extern "C" void kernel_launch(void* const* d_in, const int* in_sizes, int n_in,
                              void* d_out, int out_size, void* d_ws, size_t ws_size,
                              hipStream_t stream) {
    const float* X  = (const float*)d_in[0];   // (32,256,32,32)
    const float* E  = (const float*)d_in[1];   // (256,1024)
    const float* cs = (const float*)d_in[2];   // (1024,)
    float* out = (float*)d_out;

    char* ws = (char*)d_ws;
    int*   ws_idx   = (int*)ws;                          // 32768 ints  (128 KB)
    float* ws_enorm = (float*)(ws + 131072);             // 1024 floats (4 KB)
    float* ws_part  = (float*)(ws + 131072 + 4096);      // 4096 floats (16 KB)

    // d_out layout: [quantized_ste | vq_loss | perplexity | usage | indices(f32)]
    float* out_scalars = out + NELEM;
    float* out_idx_f   = out + NELEM + 3;

    vq_enorm_kernel<<<KCODES / 256, 256, 0, stream>>>(E, ws_enorm);
    vq_argmin_gemm_kernel<<<NROWS / 32, 256, 0, stream>>>(X, E, ws_enorm,
                                                          ws_idx, out_idx_f);
    vq_gather_kernel<<<LOSS_BLOCKS, 256, 0, stream>>>(X, E, ws_idx, out, ws_part);
    vq_finalize_kernel<<<1, 256, 0, stream>>>(ws_part, cs, out_scalars);
}